// MotifVector_62629213110678
// MI455X (gfx1250) — compile-verified
//
#include <hip/hip_runtime.h>
#include <hip/hip_bf16.h>
#include <math.h>

// ---------------- problem constants (from reference) ----------------
#define N_SAMPLES   32768
#define HDIM        1024
#define N_MOTIF     128       // 16 classes * 8 motifs
#define N_PER_CLASS 8
#define TEMPERATURE 0.2f
#define TAU         0.99f
#define K2          32
#define EPS_C       1e-4f

// GEMM tiling
#define KCHUNK      64
#define NCHUNK      (HDIM / KCHUNK)     // 16
#define KSTRIDE     68                  // 64 + 4 pad floats: bank-conflict-free, 16B-aligned rows

typedef __attribute__((ext_vector_type(2))) float v2f;
typedef __attribute__((ext_vector_type(8))) float v8f;

// ---------------- kernel 1: row sum-of-squares ----------------
__global__ void rowsumsq_kernel(const float* __restrict__ X,
                                float* __restrict__ out) {
    __shared__ float red[256];
    const float* row = X + (size_t)blockIdx.x * HDIM;
    float s = 0.f;
    for (int c = threadIdx.x; c < HDIM; c += blockDim.x) {
        float v = row[c];
        s += v * v;
    }
    red[threadIdx.x] = s;
    __syncthreads();
    for (int st = 128; st > 0; st >>= 1) {
        if (threadIdx.x < st) red[threadIdx.x] += red[threadIdx.x + st];
        __syncthreads();
    }
    if (threadIdx.x == 0) out[blockIdx.x] = red[0];
}

// ---------------- kernel 2: f32 WMMA GEMM + fused distance ----------------
// Block = 8 waves = 128 rows of z. M K-chunks are staged in LDS with
// gfx1250 async global->LDS copies (ASYNCcnt), double-buffered so the copy
// of chunk c+1 overlaps the WMMA stream on chunk c. Each wave computes a
// 16x128 strip: 8 v8f accumulators, V_WMMA_F32_16X16X4_F32 per K-step.
__global__ void __launch_bounds__(256)
gemm_dist_kernel(const float* __restrict__ z,
                 const float* __restrict__ M,
                 const float* __restrict__ zsq,
                 const float* __restrict__ msq,
                 float* __restrict__ D) {
    __shared__ float smB[2][N_MOTIF * KSTRIDE];   // 2 * 128*68*4 B = 68 KB

    const int tid  = threadIdx.x;
    const int lane = tid & 31;
    const int wave = tid >> 5;
    const int rowBase = (blockIdx.x * 8 + wave) * 16;
    const int r  = lane & 15;
    const int kp = lane >> 4;                     // 0 or 1

    // ---- async stage of one 128x64 K-chunk of M into LDS buffer (c&1) ----
    // 128 cols * 64 floats = 2048 x 16B transfers; each thread issues 8.
    auto issue_chunk = [&](int c) {
        const int kk = c * KCHUNK;
        float* buf = smB[c & 1];
#pragma unroll
        for (int it = 0; it < 8; ++it) {
            const int t   = tid + it * 256;
            const int col = t >> 4;               // 0..127
            const int seg = t & 15;               // 0..15 (16B segments)
            const float* g = M + (size_t)col * HDIM + kk + seg * 4;
            const unsigned lo =
                (unsigned)(uintptr_t)&buf[col * KSTRIDE + seg * 4];
            asm volatile("global_load_async_to_lds_b128 %0, %1, off"
                         :: "v"(lo), "v"(g) : "memory");
        }
    };

    v8f acc[8];
#pragma unroll
    for (int t = 0; t < 8; ++t) acc[t] = (v8f){0.f,0.f,0.f,0.f,0.f,0.f,0.f,0.f};

    const float* zrow = z + (size_t)(rowBase + r) * HDIM + 2 * kp;

    issue_chunk(0);
    for (int c = 0; c < NCHUNK; ++c) {
        if (c + 1 < NCHUNK) {
            issue_chunk(c + 1);
            // async loads complete in order: <=8 outstanding => chunk c landed
            asm volatile("s_wait_asynccnt 0x8" ::: "memory");
        } else {
            asm volatile("s_wait_asynccnt 0x0" ::: "memory");
        }
        __syncthreads();

        const float* bbuf = smB[c & 1];
        const int kk = c * KCHUNK;
#pragma unroll 4
        for (int k = 0; k < KCHUNK; k += 4) {
            __builtin_prefetch(zrow + kk + k + 128, 0, 1);
            v2f a = *(const v2f*)(zrow + kk + k);
#pragma unroll
            for (int t = 0; t < 8; ++t) {
                const int col = t * 16 + r;
                v2f b = *(const v2f*)(bbuf + col * KSTRIDE + k + 2 * kp);
                acc[t] = __builtin_amdgcn_wmma_f32_16x16x4_f32(
                    false, a, false, b, (short)0, acc[t], false, false);
            }
        }
        __syncthreads();   // all waves done with buf (c&1) before it is refilled
    }

    // C/D layout: VGPR v -> row v (lanes 0-15) / row v+8 (lanes 16-31), col lane&15
#pragma unroll
    for (int t = 0; t < 8; ++t) {
        const int col = t * 16 + r;
        const float mq = msq[col];
#pragma unroll
        for (int v = 0; v < 8; ++v) {
            const int row = rowBase + v + 8 * kp;
            D[(size_t)row * N_MOTIF + col] = -2.0f * acc[t][v] + zsq[row] + mq;
        }
    }
}

// ---------------- kernel 3: contrastive loss ----------------
// One wave per sample; lane owns 4 motif columns.
__global__ void loss_kernel(const float* __restrict__ D,
                            const int* __restrict__ y,
                            float* __restrict__ out_loss) {
    const int lane = threadIdx.x & 31;
    const int wave = threadIdx.x >> 5;
    const int i = blockIdx.x * (blockDim.x >> 5) + wave;   // sample index
    const int cls = y[i];

    float pos = -__builtin_huge_valf();
    float neg = 0.f;
    const float* drow = D + (size_t)i * N_MOTIF + lane * 4;
#pragma unroll
    for (int u = 0; u < 4; ++u) {
        const int col = lane * 4 + u;
        const float d = drow[u];
        const float sim = __expf(__logf((d + 1.0f) / (d + EPS_C)) / TEMPERATURE);
        if ((col >> 3) == cls) pos = fmaxf(pos, sim);
        else                   neg += sim;
    }
    // wave32 butterfly reduction
    for (int off = 16; off > 0; off >>= 1) {
        pos = fmaxf(pos, __shfl_xor(pos, off, 32));
        neg += __shfl_xor(neg, off, 32);
    }
    if (lane == 0) {
        const float term = -__logf(pos / (neg + pos));
        atomicAdd(out_loss, term * (1.0f / (float)N_SAMPLES));
    }
}

// ---------------- kernel 4: per-motif top-K2 + EMA update ----------------
// One block per motif. 32 passes of block-wide argmax with lexicographic
// (value desc, index asc) keys strictly below the previously selected key.
__global__ void __launch_bounds__(256)
topk_update_kernel(const float* __restrict__ D,
                   const int* __restrict__ y,
                   const float* __restrict__ z,
                   const float* __restrict__ Mv,
                   float* __restrict__ outM) {
    const int j   = blockIdx.x;        // motif
    const int cls = j >> 3;            // j / N_PER_CLASS

    __shared__ int   sel[K2];
    __shared__ float svals[256];
    __shared__ int   sidx[256];

    float prevV = __builtin_huge_valf();
    int   prevI = -1;

    for (int k = 0; k < K2; ++k) {
        float bestV = -__builtin_huge_valf();
        int   bestI = 0x7fffffff;
        for (int i = threadIdx.x; i < N_SAMPLES; i += blockDim.x) {
            if (y[i] != cls) continue;
            const float v = D[(size_t)i * N_MOTIF + j];
            const bool ltPrev = (v < prevV) || (v == prevV && i > prevI);
            if (!ltPrev) continue;
            const bool gtBest = (v > bestV) || (v == bestV && i < bestI);
            if (gtBest) { bestV = v; bestI = i; }
        }
        svals[threadIdx.x] = bestV;
        sidx[threadIdx.x]  = bestI;
        __syncthreads();
        for (int st = 128; st > 0; st >>= 1) {
            if (threadIdx.x < st) {
                const float v2 = svals[threadIdx.x + st];
                const int   i2 = sidx[threadIdx.x + st];
                if (v2 > svals[threadIdx.x] ||
                    (v2 == svals[threadIdx.x] && i2 < sidx[threadIdx.x])) {
                    svals[threadIdx.x] = v2;
                    sidx[threadIdx.x]  = i2;
                }
            }
            __syncthreads();
        }
        if (threadIdx.x == 0)
            sel[k] = (sidx[0] == 0x7fffffff) ? 0 : sidx[0];
        prevV = svals[0];
        prevI = sidx[0];
        __syncthreads();
    }

    // mean of selected rows + EMA
    for (int col = threadIdx.x; col < HDIM; col += blockDim.x) {
        float s = 0.f;
#pragma unroll 4
        for (int k = 0; k < K2; ++k)
            s += z[(size_t)sel[k] * HDIM + col];
        outM[(size_t)j * HDIM + col] =
            TAU * Mv[(size_t)j * HDIM + col] + (1.0f - TAU) * (s * (1.0f / K2));
    }
}

// ---------------- launch ----------------
extern "C" void kernel_launch(void* const* d_in, const int* in_sizes, int n_in,
                              void* d_out, int out_size, void* d_ws, size_t ws_size,
                              hipStream_t stream) {
    const float* z  = (const float*)d_in[0];   // [32768,1024]
    const float* Mv = (const float*)d_in[1];   // [128,1024]
    const int*   y  = (const int*)d_in[2];     // [32768]

    float* out  = (float*)d_out;               // [0]=loss, [1..]=new_M (128*1024)
    float* D    = (float*)d_ws;                                // 32768*128 f32
    float* zsq  = D + (size_t)N_SAMPLES * N_MOTIF;             // 32768 f32
    float* msq  = zsq + N_SAMPLES;                             // 128 f32

    // row norms
    rowsumsq_kernel<<<N_SAMPLES, 256, 0, stream>>>(z, zsq);
    rowsumsq_kernel<<<N_MOTIF,   256, 0, stream>>>(Mv, msq);

    // WMMA GEMM + distance: 2048 waves, 8 waves/block -> 256 blocks
    gemm_dist_kernel<<<N_SAMPLES / 16 / 8, 256, 0, stream>>>(z, Mv, zsq, msq, D);

    // loss: zero accumulator then reduce (one wave per sample)
    hipMemsetAsync(d_out, 0, sizeof(float), stream);
    loss_kernel<<<N_SAMPLES / 8, 256, 0, stream>>>(D, y, out);

    // top-K2 + EMA motif update
    topk_update_kernel<<<N_MOTIF, 256, 0, stream>>>(D, y, z, Mv, out + 1);
}